// SpatialExt_47991964566190
// MI455X (gfx1250) — compile-verified
//
#include <hip/hip_runtime.h>

// MI455X / gfx1250 fused HGNN kernel.
// One workgroup (128 threads = 4 wave32) per (b,f) pair; loops s=0..2,
// builds G with V_WMMA_F32_16X16X4_F32 (full fp32 precision), keeps all
// intermediates in LDS, then does the [62,57]@[57,256] embedding GEMM with
// the same f32 WMMA. Bandwidth-bound (449 MB output); each byte written once.

typedef float v2f __attribute__((ext_vector_type(2)));
typedef float v8f __attribute__((ext_vector_type(8)));

namespace {
constexpr int kS    = 3;    // SEQ_LEN
constexpr int kFeat = 5;    // FEAT (EAdj rows)
constexpr int kE    = 5;    // EMBED
constexpr int kHid  = 6;
constexpr int kCls  = 8;
constexpr int kB    = 32;
constexpr int kF    = 128;
constexpr int kC    = 62;   // CHANNEL == HELEN
constexpr int kOut  = 256;
constexpr int CP    = 64;                       // C padded to 4x 16-tiles
constexpr int KCAT  = (kE + kHid + kCls) * kS;  // 57
}

__global__ __launch_bounds__(128)
void hgnn_fused_kernel(const float* __restrict__ x,     // [B,S,F,C,E]
                       const float* __restrict__ EAdj,  // [FEAT, S*C]
                       const float* __restrict__ W1,    // [E,HID]
                       const float* __restrict__ b1,    // [HID]
                       const float* __restrict__ W2,    // [HID,CLS]
                       const float* __restrict__ b2,    // [CLS]
                       const float* __restrict__ Wemb,  // [57,256]
                       const float* __restrict__ bemb,  // [256]
                       float* __restrict__ out,         // [B,F,C,256]
                       float* __restrict__ Gout)        // [B,S,F,C,C]
{
    __shared__ float Hs[CP][CP];        // soft incidence, zero padded
    __shared__ float Gs[CP][CP];        // normalized Laplacian tile
    __shared__ float featA[CP][CP];     // [c][s*19+j] concat features, padded
    __shared__ float xs[kC][kE];
    __shared__ float an[kC][kE];
    __shared__ float cw[CP];            // w[e] / De[e]  (column scale for M)
    __shared__ float isdv[CP];          // Dv^{-1/2}
    __shared__ float wloc[CP];
    __shared__ float P1[kC][kHid], h1s[kC][kHid];
    __shared__ float P2[kC][kCls], h2s[kC][kCls];
    __shared__ float W1s[kE * kHid], W2s[kHid * kCls], b1s[kHid], b2s[kCls];

    const int tid   = threadIdx.x;
    const int lane  = tid & 31;
    const int wave  = tid >> 5;
    const int half2 = (lane < 16) ? 0 : 2;   // 32-bit A/B frag K split
    const int half8 = (lane < 16) ? 0 : 8;   // 32-bit C/D frag row split
    const int l16   = lane & 15;

    const int bf = blockIdx.x;
    const int b  = bf / kF;
    const int f  = bf % kF;

    for (int i = tid; i < CP * CP; i += 128) (&featA[0][0])[i] = 0.f;
    if (tid < kE * kHid)   W1s[tid] = W1[tid];
    if (tid < kHid)        b1s[tid] = b1[tid];
    if (tid < kHid * kCls) W2s[tid] = W2[tid];
    if (tid < kCls)        b2s[tid] = b2[tid];

    for (int s = 0; s < kS; ++s) {
        const int slice = (b * kS + s) * kF + f;
        const float* xp = x + (size_t)slice * kC * kE;

        __syncthreads();  // previous iteration's consumers of xs/Hs/Gs done
        for (int i = tid; i < kC * kE; i += 128) (&xs[0][0])[i] = xp[i];
        if (tid < CP) {   // hyperedge weights: mean over FEAT of EAdj
            float w = 0.f;
            if (tid < kC) {
                for (int ff = 0; ff < kFeat; ++ff)
                    w += EAdj[ff * (kS * kC) + s * kC + tid];
                w *= (1.f / kFeat);
            }
            wloc[tid] = w;
        }
        __syncthreads();

        if (tid < kC) {   // row-normalize for cosine similarity
            float n2 = 0.f;
            for (int e = 0; e < kE; ++e) n2 += xs[tid][e] * xs[tid][e];
            const float inv = 1.f / fmaxf(sqrtf(n2), 1e-8f);
            for (int e = 0; e < kE; ++e) an[tid][e] = xs[tid][e] * inv;
        }
        __syncthreads();

        // H = (cos_sim + 1)/2, zero padded to 64x64
        for (int i = tid; i < CP * CP; i += 128) {
            const int c = i >> 6, d = i & 63;
            float h = 0.f;
            if (c < kC && d < kC) {
                float acc = 0.f;
                for (int e = 0; e < kE; ++e) acc += an[c][e] * an[d][e];
                h = (acc + 1.f) * 0.5f;
            }
            Hs[c][d] = h;
        }
        __syncthreads();

        if (tid < CP) {   // edge/node degrees
            if (tid < kC) {
                float de = 0.f, dv = 0.f;
                for (int c = 0; c < kC; ++c) de += Hs[c][tid];
                for (int e = 0; e < kC; ++e) dv += Hs[tid][e] * wloc[e];
                cw[tid]   = wloc[tid] / fmaxf(de, 1e-8f);
                isdv[tid] = 1.f / sqrtf(fmaxf(dv, 1e-8f));
            } else {
                cw[tid] = 0.f; isdv[tid] = 0.f;
            }
        }
        __syncthreads();

        // G = (H * diag(cw)) @ H^T via fp32 WMMA; wave owns column tile `wave`
        {
            const int nt = wave;
            for (int mt = 0; mt < 4; ++mt) {
                v8f acc = {};
                for (int k = 0; k < CP; k += 4) {
                    const int k0 = k + half2;
                    v2f a, bb;
                    a[0]  = Hs[mt * 16 + l16][k0]     * cw[k0];
                    a[1]  = Hs[mt * 16 + l16][k0 + 1] * cw[k0 + 1];
                    bb[0] = Hs[nt * 16 + l16][k0];        // B[k][d] = H[d][k]
                    bb[1] = Hs[nt * 16 + l16][k0 + 1];
                    acc = __builtin_amdgcn_wmma_f32_16x16x4_f32(
                        false, a, false, bb, (short)0, acc, false, false);
                }
#pragma unroll
                for (int v = 0; v < 8; ++v) {
                    const int c = mt * 16 + v + half8;
                    const int d = nt * 16 + l16;
                    const float g = acc[v] * isdv[c] * isdv[d];
                    Gs[c][d] = g;
                    if (c < kC && d < kC)
                        Gout[((size_t)slice * kC + c) * kC + d] = g;
                }
            }
        }
        __syncthreads();

        // P1 = x @ W1 (62x5 @ 5x6)
        for (int i = tid; i < kC * kHid; i += 128) {
            const int c = i / kHid, j = i % kHid;
            float acc = 0.f;
            for (int e = 0; e < kE; ++e) acc += xs[c][e] * W1s[e * kHid + j];
            P1[c][j] = acc;
        }
        __syncthreads();
        // h1 = relu(G @ P1 + b1)
        for (int i = tid; i < kC * kHid; i += 128) {
            const int c = i / kHid, j = i % kHid;
            float acc = b1s[j];
            for (int d = 0; d < kC; ++d) acc += Gs[c][d] * P1[d][j];
            h1s[c][j] = fmaxf(acc, 0.f);
        }
        __syncthreads();
        // P2 = h1 @ W2 (62x6 @ 6x8)
        for (int i = tid; i < kC * kCls; i += 128) {
            const int c = i / kCls, j = i % kCls;
            float acc = 0.f;
            for (int kk = 0; kk < kHid; ++kk) acc += h1s[c][kk] * W2s[kk * kCls + j];
            P2[c][j] = acc;
        }
        __syncthreads();
        // h2 = G @ P2 + b2
        for (int i = tid; i < kC * kCls; i += 128) {
            const int c = i / kCls, j = i % kCls;
            float acc = b2s[j];
            for (int d = 0; d < kC; ++d) acc += Gs[c][d] * P2[d][j];
            h2s[c][j] = acc;
        }
        __syncthreads();
        // pack concat([x, h1, h2]) into padded feature tile at [c][s*19+j]
        for (int i = tid; i < kC * 19; i += 128) {
            const int c = i / 19, j = i % 19;
            const float v = (j < kE) ? xs[c][j]
                          : (j < kE + kHid) ? h1s[c][j - kE]
                                            : h2s[c][j - kE - kHid];
            featA[c][s * 19 + j] = v;
        }
    }
    __syncthreads();

    // out[b,f] = featA(62x57,padded 64x64) @ Wemb(57x256) + bemb, fp32 WMMA.
    // 16 N-tiles split over 4 waves; B fragments hoisted and reused over M-tiles.
    for (int nt = wave; nt < 16; nt += 4) {
        const int n = nt * 16 + l16;
        v2f bfr[16];
#pragma unroll
        for (int kk = 0; kk < 16; ++kk) {
            const int k0 = kk * 4 + half2;
            bfr[kk][0] = (k0     < KCAT) ? Wemb[(k0)     * kOut + n] : 0.f;
            bfr[kk][1] = (k0 + 1 < KCAT) ? Wemb[(k0 + 1) * kOut + n] : 0.f;
        }
        const float bias = bemb[n];
        for (int mt = 0; mt < 4; ++mt) {
            v8f acc = {};
#pragma unroll
            for (int kk = 0; kk < 16; ++kk) {
                const int k0 = kk * 4 + half2;
                v2f a;
                a[0] = featA[mt * 16 + l16][k0];
                a[1] = featA[mt * 16 + l16][k0 + 1];
                acc = __builtin_amdgcn_wmma_f32_16x16x4_f32(
                    false, a, false, bfr[kk], (short)0, acc, false, false);
            }
#pragma unroll
            for (int v = 0; v < 8; ++v) {
                const int c = mt * 16 + v + half8;
                if (c < kC)
                    out[(((size_t)b * kF + f) * kC + c) * kOut + n] = acc[v] + bias;
            }
        }
    }
}

extern "C" void kernel_launch(void* const* d_in, const int* in_sizes, int n_in,
                              void* d_out, int out_size, void* d_ws, size_t ws_size,
                              hipStream_t stream) {
    const float* x    = (const float*)d_in[0];
    const float* EAdj = (const float*)d_in[1];
    const float* W1   = (const float*)d_in[2];
    const float* b1   = (const float*)d_in[3];
    const float* W2   = (const float*)d_in[4];
    const float* b2   = (const float*)d_in[5];
    const float* Wemb = (const float*)d_in[6];
    const float* bemb = (const float*)d_in[7];
    (void)in_sizes; (void)n_in; (void)d_ws; (void)ws_size; (void)out_size;

    float* out  = (float*)d_out;                        // [B,F,C,256]
    float* Gout = out + (size_t)kB * kF * kC * kOut;    // [B,S,F,C,C]

    hgnn_fused_kernel<<<kB * kF, 128, 0, stream>>>(
        x, EAdj, W1, b1, W2, b2, Wemb, bemb, out, Gout);
}